// RelPositionMultiheadAttentionWeights_62268435858106
// MI455X (gfx1250) — compile-verified
//
#include <hip/hip_runtime.h>
#include <hip/hip_bf16.h>

typedef __attribute__((ext_vector_type(2))) float v2f;
typedef __attribute__((ext_vector_type(8))) float v8f;

#define S_LEN  1024
#define BATCH  8
#define EMB    512
#define PDIM   192
#define NH     8
#define QHD    32
#define PHD    4
#define IN_DIM 544            // (2*QHD + PHD) * NH
#define NPOS   2047           // 2*S - 1
#define PE_ROWS (BATCH * NPOS)   // 16376
#define PROJ_ELEMS ((size_t)8192 * IN_DIM)

__device__ __forceinline__ v8f wmma_f32_k4(v2f a, v2f b, v8f c) {
  // D = A(16x4 f32) * B(4x16 f32) + C(16x16 f32)
  return __builtin_amdgcn_wmma_f32_16x16x4_f32(
      /*neg_a=*/false, a, /*neg_b=*/false, b,
      /*c_mod=*/(short)0, c, /*reuse_a=*/false, /*reuse_b=*/false);
}

// ---------------------------------------------------------------------------
// Kernel 1: proj = x @ W_in + b_in        x:[8192,512]  W_in:[512,544]
// One 16x16 tile per wave, K-loop of 128 f32 WMMA (K=4) steps.
// grid = (34 n-tiles, 64 m-blocks of 8 waves), block = 256
// ---------------------------------------------------------------------------
__global__ __launch_bounds__(256) void in_proj_kernel(
    const float* __restrict__ x, const float* __restrict__ W_in,
    const float* __restrict__ b_in, float* __restrict__ proj) {
  const int wave = threadIdx.x >> 5;
  const int lane = threadIdx.x & 31;
  const int lo = lane & 15, hi = lane >> 4, khalf = hi * 2;
  const int mt = blockIdx.y * 8 + wave;     // 0..511
  const int nt = blockIdx.x;                // 0..33
  const int arow = mt * 16 + lo;
  const int col  = nt * 16 + lo;
  const float* A = x + (size_t)arow * EMB;
  v8f c = {};
  #pragma unroll 4
  for (int k0 = 0; k0 < EMB; k0 += 4) {
    v2f a, b;
    a.x = A[k0 + khalf];
    a.y = A[k0 + khalf + 1];
    b.x = W_in[(size_t)(k0 + khalf)     * IN_DIM + col];
    b.y = W_in[(size_t)(k0 + khalf + 1) * IN_DIM + col];
    c = wmma_f32_k4(a, b, c);
  }
  const float bias = b_in[col];
  #pragma unroll
  for (int v = 0; v < 8; ++v) {
    const int r = mt * 16 + v + 8 * hi;     // C/D layout: row = v + 8*hi
    proj[(size_t)r * IN_DIM + col] = c[v] + bias;
  }
}

// ---------------------------------------------------------------------------
// Kernel 2: pe = pos_emb @ W_pos          pos_emb:[16376,192]  W_pos:[192,32]
// 1024 m-tiles x 2 n-tiles = 2048 wave-tiles; ragged edge rows clamped on
// load, guarded on store (after all WMMAs -> EXEC all-1s during WMMA).
// grid = 256 blocks, block = 256
// ---------------------------------------------------------------------------
__global__ __launch_bounds__(256) void pos_proj_kernel(
    const float* __restrict__ pos_emb, const float* __restrict__ W_pos,
    float* __restrict__ pe) {
  const int wave = threadIdx.x >> 5;
  const int lane = threadIdx.x & 31;
  const int lo = lane & 15, hi = lane >> 4, khalf = hi * 2;
  const int tile = blockIdx.x * 8 + wave;   // 0..2047
  const int mt = tile >> 1;
  const int nt = tile & 1;
  int arow = mt * 16 + lo;
  if (arow >= PE_ROWS) arow = PE_ROWS - 1;  // clamp, never diverge at WMMA
  const int col = nt * 16 + lo;
  const float* A = pos_emb + (size_t)arow * PDIM;
  v8f c = {};
  #pragma unroll 4
  for (int k0 = 0; k0 < PDIM; k0 += 4) {
    v2f a, b;
    a.x = A[k0 + khalf];
    a.y = A[k0 + khalf + 1];
    b.x = W_pos[(k0 + khalf)     * (NH * PHD) + col];
    b.y = W_pos[(k0 + khalf + 1) * (NH * PHD) + col];
    c = wmma_f32_k4(a, b, c);
  }
  #pragma unroll
  for (int v = 0; v < 8; ++v) {
    const int r = mt * 16 + v + 8 * hi;
    if (r < PE_ROWS) pe[(size_t)r * (NH * PHD) + col] = c[v];
  }
}

// ---------------------------------------------------------------------------
// Kernel 3: scores = softmax(Q.K^T + skew(P.PE^T)) , out:[H,B,S,S]
// grid = (S/16 row-tiles, H*B), block = 256 (8 waves).
// Wave w holds cols [128w,128w+128) as 8 WMMA C tiles; rel added as per-
// element float4 dots (skewed gather); softmax via shfl_xor + LDS.
// ---------------------------------------------------------------------------
__global__ __launch_bounds__(256) void attn_softmax_kernel(
    const float* __restrict__ proj, const float* __restrict__ pe,
    float* __restrict__ out) {
  __shared__ float red_max[8][16];
  __shared__ float red_sum[8][16];
  const int wave = threadIdx.x >> 5;
  const int lane = threadIdx.x & 31;
  const int lo = lane & 15, hi = lane >> 4, khalf = hi * 2;
  const int hb = blockIdx.y;                // h*BATCH + b
  const int h = hb >> 3, b = hb & 7;
  const int i0 = blockIdx.x * 16;

  // Preload Q A-operands (16 VGPRs): rows i0+lo, dims [h*32, h*32+32)
  const float* Qrow = proj + ((size_t)(i0 + lo) * BATCH + b) * IN_DIM + h * QHD;
  v2f qa[8];
  #pragma unroll
  for (int kk = 0; kk < 8; ++kk) {
    qa[kk].x = Qrow[kk * 4 + khalf];
    qa[kk].y = Qrow[kk * 4 + khalf + 1];
  }

  // Q.K^T: 8 column tiles x 8 K=4 WMMA steps
  v8f acc[8];
  #pragma unroll
  for (int t = 0; t < 8; ++t) {
    const int j = wave * 128 + t * 16 + lo;
    const float* Krow =
        proj + ((size_t)j * BATCH + b) * IN_DIM + NH * QHD + h * QHD;
    v8f c = {};
    #pragma unroll
    for (int kk = 0; kk < 8; ++kk) {
      v2f kb;
      kb.x = Krow[kk * 4 + khalf];
      kb.y = Krow[kk * 4 + khalf + 1];
      c = wmma_f32_k4(qa[kk], kb, c);
    }
    acc[t] = c;
  }

  // rel[i,j] = P[i,:] . PE[b, S-1-i+j, h, :]   (n always in [0, 2046])
  #pragma unroll
  for (int v = 0; v < 8; ++v) {
    const int srow = i0 + v + 8 * hi;
    const float4 p4 = *(const float4*)(
        proj + ((size_t)srow * BATCH + b) * IN_DIM + 2 * NH * QHD + h * PHD);
    #pragma unroll
    for (int t = 0; t < 8; ++t) {
      const int j = wave * 128 + t * 16 + lo;
      const int n = (S_LEN - 1) - srow + j;
      const float4 e4 = *(const float4*)(
          pe + ((size_t)(b * NPOS + n)) * (NH * PHD) + h * PHD);
      acc[t][v] += p4.x * e4.x + p4.y * e4.y + p4.z * e4.z + p4.w * e4.w;
    }
  }

  // ---- row max: per-lane over tiles -> 16-lane half shuffle -> LDS x-wave
  float m[8];
  #pragma unroll
  for (int v = 0; v < 8; ++v) {
    float mv = acc[0][v];
    #pragma unroll
    for (int t = 1; t < 8; ++t) mv = fmaxf(mv, acc[t][v]);
    #pragma unroll
    for (int off = 1; off < 16; off <<= 1)
      mv = fmaxf(mv, __shfl_xor(mv, off, 32));
    m[v] = mv;
  }
  if (lo == 0) {
    #pragma unroll
    for (int v = 0; v < 8; ++v) red_max[wave][v + 8 * hi] = m[v];
  }
  __syncthreads();
  #pragma unroll
  for (int v = 0; v < 8; ++v) {
    float g = red_max[0][v + 8 * hi];
    #pragma unroll
    for (int ww = 1; ww < 8; ++ww) g = fmaxf(g, red_max[ww][v + 8 * hi]);
    m[v] = g;
  }

  // ---- exp + row sum
  float s[8];
  #pragma unroll
  for (int v = 0; v < 8; ++v) {
    float sv = 0.f;
    #pragma unroll
    for (int t = 0; t < 8; ++t) {
      const float e = __expf(acc[t][v] - m[v]);
      acc[t][v] = e;
      sv += e;
    }
    #pragma unroll
    for (int off = 1; off < 16; off <<= 1)
      sv += __shfl_xor(sv, off, 32);
    s[v] = sv;
  }
  if (lo == 0) {
    #pragma unroll
    for (int v = 0; v < 8; ++v) red_sum[wave][v + 8 * hi] = s[v];
  }
  __syncthreads();
  #pragma unroll
  for (int v = 0; v < 8; ++v) {
    float g = 0.f;
    #pragma unroll
    for (int ww = 0; ww < 8; ++ww) g += red_sum[ww][v + 8 * hi];
    s[v] = 1.0f / g;
  }

  // ---- normalize + store (out[(h*B+b), i, j])
  float* obase = out + (size_t)hb * S_LEN * S_LEN;
  #pragma unroll
  for (int v = 0; v < 8; ++v) {
    const int r = i0 + v + 8 * hi;
    #pragma unroll
    for (int t = 0; t < 8; ++t) {
      const int j = wave * 128 + t * 16 + lo;
      obase[(size_t)r * S_LEN + j] = acc[t][v] * s[v];
    }
  }
}

extern "C" void kernel_launch(void* const* d_in, const int* in_sizes, int n_in,
                              void* d_out, int out_size, void* d_ws, size_t ws_size,
                              hipStream_t stream) {
  const float* x     = (const float*)d_in[0];   // [S,B,E]
  const float* pos   = (const float*)d_in[1];   // [B,2S-1,PD]
  const float* W_in  = (const float*)d_in[2];   // [E,544]
  const float* b_in  = (const float*)d_in[3];   // [544]
  const float* W_pos = (const float*)d_in[4];   // [PD,32]

  float* proj = (float*)d_ws;                   // [8192, 544]
  float* pe   = proj + PROJ_ELEMS;              // [16376, 32]
  float* outp = (float*)d_out;                  // [H,B,S,S]

  in_proj_kernel<<<dim3(IN_DIM / 16, 64), 256, 0, stream>>>(x, W_in, b_in, proj);
  pos_proj_kernel<<<dim3(256), 256, 0, stream>>>(pos, W_pos, pe);
  attn_softmax_kernel<<<dim3(S_LEN / 16, NH * BATCH), 256, 0, stream>>>(proj, pe, outp);
}